// Graph_ReLu_W_18734647345753
// MI455X (gfx1250) — compile-verified
//
#include <hip/hip_runtime.h>
#include <stdint.h>

// Problem constants (fixed by the reference).
#define NROW 8192   // row length == number of rows
#define KSEL 32     // top-k per row
#define NT   256    // threads per block = 8 wave32 waves
#define EPT  (NROW / NT)   // 32 elements per thread

typedef uint32_t u32x4 __attribute__((ext_vector_type(4)));
typedef int32_t  i32x8 __attribute__((ext_vector_type(8)));
typedef int32_t  i32x4 __attribute__((ext_vector_type(4)));

// ---- Tensor Data Mover descriptor for one contiguous 8192-float row ----
// D# group0 (128b): count=1, lds_addr[63:32], global_addr[120:64], type=2[127:126]
// D# group1 (256b): data_size=4B, tensor_dim0=8192, tensor_dim1=1,
//                   tile_dim0=8192, tile_dim1=1, tensor_dim0_stride=8192
__device__ __forceinline__ void make_row_desc(uint64_t gaddr, uint32_t laddr,
                                              u32x4 &g0, i32x8 &g1) {
  g0.x = 1u;                                     // count = 1 (valid descriptor)
  g0.y = laddr;                                  // lds_addr (bytes)
  g0.z = (uint32_t)gaddr;                        // global_addr[31:0]
  g0.w = (uint32_t)(gaddr >> 32) | (2u << 30);   // global_addr[56:32] | type=2
  g1[0] = (int)(2u << 16);        // workgroup_mask=0, data_size=2 (4 bytes)
  g1[1] = (int)((uint32_t)NROW << 16); // tensor_dim0[15:0] at bits 63:48
  g1[2] = (int)(1u << 16);        // tensor_dim0 hi=0; tensor_dim1[15:0]=1 at 95:80
  g1[3] = (int)((uint32_t)NROW << 16); // tensor_dim1 hi=0; tile_dim0=8192 at 127:112
  g1[4] = 1;                      // tile_dim1=1 at 143:128; tile_dim2=0
  g1[5] = NROW;                   // tensor_dim0_stride[31:0] at 191:160
  g1[6] = 0;                      // stride hi, tensor_dim1_stride lo
  g1[7] = 0;
}

__device__ __forceinline__ void tdm_load_row(uint64_t gaddr, uint32_t laddr) {
  u32x4 g0; i32x8 g1; make_row_desc(gaddr, laddr, g0, g1);
  i32x4 z4 = {0, 0, 0, 0};
#if __clang_major__ >= 23
  i32x8 z8 = {0, 0, 0, 0, 0, 0, 0, 0};
  __builtin_amdgcn_tensor_load_to_lds(g0, g1, z4, z4, z8, 0);
#else
  __builtin_amdgcn_tensor_load_to_lds(g0, g1, z4, z4, 0);
#endif
}

__device__ __forceinline__ void tdm_store_row(uint64_t gaddr, uint32_t laddr) {
  u32x4 g0; i32x8 g1; make_row_desc(gaddr, laddr, g0, g1);
  i32x4 z4 = {0, 0, 0, 0};
#if __clang_major__ >= 23
  i32x8 z8 = {0, 0, 0, 0, 0, 0, 0, 0};
  __builtin_amdgcn_tensor_store_from_lds(g0, g1, z4, z4, z8, 0);
#else
  __builtin_amdgcn_tensor_store_from_lds(g0, g1, z4, z4, 0);
#endif
}

// One workgroup per row: TDM-stage row in LDS, radix-select the K-th largest
// relu'd value (fp32 bits of non-negative floats are order-isomorphic to u32),
// keep exact lowest-index tie handling, write masked row back via TDM.
extern "C" __global__ void __launch_bounds__(NT)
graph_relu_topk_kernel(const float* __restrict__ A, float* __restrict__ out) {
  __shared__ __align__(16) float buf[NROW];     // 32 KB row buffer
  __shared__ uint32_t hist[256];
  __shared__ uint32_t scanb[NT];                // reused: bin suffix-scan & tie scan
  __shared__ uint32_t sel[2];                   // [0]=selected bin, [1]=residual need
  __shared__ uint32_t kept_idx[KSEL];           // indices of kept tie elements
  __shared__ uint32_t kept_n;

  const int t = threadIdx.x;
  const uint64_t byte_off = (uint64_t)blockIdx.x * (uint64_t)NROW * 4ull;

  // --- Stage row into LDS via Tensor Data Mover (wave 0 only; EXEC-uniform) ---
  if (t < 32) {
    tdm_load_row((uint64_t)(uintptr_t)A + byte_off, (uint32_t)(uintptr_t)&buf[0]);
    __builtin_amdgcn_s_wait_tensorcnt(0);
  }
  __syncthreads();

  // --- In-place ReLU (lane-contiguous float4, bank-conflict free) ---
  for (int i = 0; i < EPT / 4; ++i) {
    const int base = (i * NT + t) * 4;
    float4 v = *reinterpret_cast<float4*>(&buf[base]);
    v.x = fmaxf(v.x, 0.0f); v.y = fmaxf(v.y, 0.0f);
    v.z = fmaxf(v.z, 0.0f); v.w = fmaxf(v.w, 0.0f);
    *reinterpret_cast<float4*>(&buf[base]) = v;
  }
  __syncthreads();

  // --- 4-pass 8-bit radix select for the K-th largest bit pattern ---
  uint32_t prefix = 0, need = KSEL;
  for (int pass = 0; pass < 4; ++pass) {
    const int shift = 24 - 8 * pass;
    const uint32_t pmask = pass ? (0xFFFFFFFFu << (shift + 8)) : 0u;

    hist[t] = 0;                 // NT == 256 bins
    __syncthreads();
    for (int i = 0; i < EPT; ++i) {
      const uint32_t u = __float_as_uint(buf[i * NT + t]);
      if ((u & pmask) == prefix)
        atomicAdd(&hist[(u >> shift) & 0xFFu], 1u);
    }
    __syncthreads();

    // Parallel suffix sum over bins (high bin = larger value).
    scanb[t] = hist[t];
    __syncthreads();
    for (int off = 1; off < NT; off <<= 1) {
      const uint32_t mine = scanb[t];
      const uint32_t add  = (t + off < NT) ? scanb[t + off] : 0u;
      __syncthreads();
      scanb[t] = mine + add;
      __syncthreads();
    }
    {
      const uint32_t incl = scanb[t];          // count in bins >= t
      const uint32_t excl = incl - hist[t];    // count in bins >  t
      if (incl >= need && excl < need) {       // exactly one bin qualifies
        sel[0] = (uint32_t)t;
        sel[1] = need - excl;                  // residual rank inside this bin
      }
    }
    __syncthreads();
    prefix |= sel[0] << shift;
    need = sel[1];
    __syncthreads();
  }

  const uint32_t thr = prefix;   // bit pattern of the K-th largest relu value
  const uint32_t T   = need;     // how many elements equal to thr to keep
  if (t == 0) kept_n = 0;
  __syncthreads();

  // --- Exact tie handling: keep first T equals in ascending index order.
  // (Only output-relevant when thr > 0; for thr == 0 kept values are 0 anyway.)
  if (thr != 0u) {
    uint32_t local_eq = 0;
    for (int i = 0; i < EPT; ++i)
      if (__float_as_uint(buf[t * EPT + i]) == thr) local_eq++;
    scanb[t] = local_eq;
    __syncthreads();
    for (int off = 1; off < NT; off <<= 1) {   // inclusive Hillis-Steele scan
      const uint32_t mine = scanb[t];
      const uint32_t add  = (t >= off) ? scanb[t - off] : 0u;
      __syncthreads();
      scanb[t] = mine + add;
      __syncthreads();
    }
    uint32_t r = (t == 0) ? 0u : scanb[t - 1]; // exclusive rank of my first equal
    for (int i = 0; i < EPT && r < T; ++i) {
      if (__float_as_uint(buf[t * EPT + i]) == thr) {
        kept_idx[r] = (uint32_t)(t * EPT + i);
        r++;
      }
    }
    if (t == 0) {
      const uint32_t tot = scanb[NT - 1];
      kept_n = (tot < T) ? tot : T;
    }
  }
  __syncthreads();
  const uint32_t nk = kept_n;

  // --- Apply mask in place: keep if bits > thr, or an index-kept tie ---
  for (int i = 0; i < EPT / 4; ++i) {
    const int base = (i * NT + t) * 4;
    float4 v = *reinterpret_cast<float4*>(&buf[base]);
    float o[4] = {v.x, v.y, v.z, v.w};
#pragma unroll
    for (int j = 0; j < 4; ++j) {
      const uint32_t u = __float_as_uint(o[j]);
      bool keep = (u > thr);                   // unsigned cmp == float cmp (>=0)
      if (!keep && u == thr && thr != 0u) {
        const uint32_t e = (uint32_t)(base + j);
        for (uint32_t q = 0; q < nk; ++q) keep = keep || (kept_idx[q] == e);
      }
      if (!keep) o[j] = 0.0f;
    }
    v.x = o[0]; v.y = o[1]; v.z = o[2]; v.w = o[3];
    *reinterpret_cast<float4*>(&buf[base]) = v;
  }
  __syncthreads();

  // --- Stream masked row back out via Tensor Data Mover (wave 0 only) ---
  if (t < 32) {
    tdm_store_row((uint64_t)(uintptr_t)out + byte_off, (uint32_t)(uintptr_t)&buf[0]);
    __builtin_amdgcn_s_wait_tensorcnt(0);
  }
}

extern "C" void kernel_launch(void* const* d_in, const int* in_sizes, int n_in,
                              void* d_out, int out_size, void* d_ws, size_t ws_size,
                              hipStream_t stream) {
  (void)in_sizes; (void)n_in; (void)d_ws; (void)ws_size; (void)out_size;
  const float* A = (const float*)d_in[0];   // [8192, 8192] fp32
  float* out = (float*)d_out;               // [8192, 8192] fp32
  graph_relu_topk_kernel<<<dim3(NROW), dim3(NT), 0, stream>>>(A, out);
}